// RWKV_RNN_49031346651566
// MI455X (gfx1250) — compile-verified
//
// RWKV single-token inference for MI455X (gfx1250), single persistent kernel.
//
// Roofline: 1.46 GB of fp32 weights streamed once -> ~63 us floor at 23.3 TB/s.
// Latency-bound otherwise (~150 dependent steps), so one persistent kernel with
// atomic grid barriers instead of ~150 kernel launches. Matvecs run on
// v_wmma_f32_16x16x4_f32 (A = weight tile, B = x broadcast across columns),
// weights loaded with non-temporal b128 (read-once, > L2 capacity).

#include <hip/hip_runtime.h>
#include <cstdint>

#define Edim 1024
#define Hdim 4096
#define Ldim 24
#define Vdim 50277
#define NWG  512
#define NTH  256
#define NWAVE 8

typedef float v2f __attribute__((ext_vector_type(2)));
typedef float v4f __attribute__((ext_vector_type(4)));
typedef float v8f __attribute__((ext_vector_type(8)));

struct Params {
  const int* ctx;
  const float *xx_att, *aa_att, *bb_att, *pp_att, *xx_ffn;
  const float *emb_w, *head_w;
  const float *ln0_w, *ln0_b, *ln1_w, *ln1_b, *ln2_w, *ln2_b, *lno_w, *lno_b;
  const float *mix_k, *mix_v, *mix_r, *tf, *td;
  const float *Wr, *Wk, *Wv, *Wo;
  const float *fmk, *fmr, *Fr, *Fk, *Fv;
  float* out;
  int*   cnt;   // grid barrier counter (zeroed per call via hipMemsetAsync)
  float* ws;    // float scratch
};

// ---------------- grid barrier (monotonic epoch counter) ----------------
__device__ __forceinline__ void gbar(int* cnt, int epoch) {
  __threadfence();                 // make our global stores device-visible
  __syncthreads();
  if (threadIdx.x == 0) {
    __hip_atomic_fetch_add(cnt, 1, __ATOMIC_RELEASE, __HIP_MEMORY_SCOPE_AGENT);
    while (__hip_atomic_load(cnt, __ATOMIC_ACQUIRE, __HIP_MEMORY_SCOPE_AGENT)
           < epoch * NWG) {
      __builtin_amdgcn_s_sleep(2);
    }
  }
  __syncthreads();
}

// ---------------- WMMA f32 16x16x4 wrapper ----------------
__device__ __forceinline__ v8f wmma4(v2f a, v2f b, v8f c) {
  // (neg_a, A, neg_b, B, c_mod, C, reuse_a, reuse_b)
  return __builtin_amdgcn_wmma_f32_16x16x4_f32(false, a, false, b, (short)0, c,
                                               false, false);
}

__device__ __forceinline__ v4f ld_nt(const float* p) {
  return __builtin_nontemporal_load((const v4f*)p);   // global_load_b128 nt
}

// One wave: 16-row strip of W (ldw = cols), contract k in [k0, k0+klen),
// klen a multiple of 16. A-layout: lanes 0-15 hold K pair (k,k+1), lanes
// 16-31 hold (k+2,k+3) -> with a per-lane float4 we remap the 4 hardware K
// slots to logical ks {k,k+1 | k+4,k+5} and {k+2,k+3 | k+6,k+7}; B (the
// broadcast x) uses the identical mapping, so the contraction is exact.
__device__ __forceinline__ v8f mv_wave(const float* __restrict__ W,
                                       const float* __restrict__ x,
                                       int ldw, int row0, int rowmax,
                                       int k0, int klen) {
  const int lane = threadIdx.x & 31;
  const int half = lane >> 4;
  int r = row0 + (lane & 15);
  if (r > rowmax) r = rowmax;                    // clamp tail loads (head matvec)
  const float* wp = W + (size_t)r * ldw + (k0 + 4 * half);
  const float* xp = x + (k0 + 4 * half);
  v8f acc = {0.f, 0.f, 0.f, 0.f, 0.f, 0.f, 0.f, 0.f};
  for (int k = 0; k < klen; k += 16) {
    v4f w0 = ld_nt(wp + k);
    v4f w1 = ld_nt(wp + k + 8);
    v4f x0 = *(const v4f*)(xp + k);
    v4f x1 = *(const v4f*)(xp + k + 8);
    v2f a0 = {w0.x, w0.y}, b0 = {x0.x, x0.y};
    v2f a1 = {w0.z, w0.w}, b1 = {x0.z, x0.w};
    v2f a2 = {w1.x, w1.y}, b2 = {x1.x, x1.y};
    v2f a3 = {w1.z, w1.w}, b3 = {x1.z, x1.w};
    acc = wmma4(a0, b0, acc);
    acc = wmma4(a1, b1, acc);
    acc = wmma4(a2, b2, acc);
    acc = wmma4(a3, b3, acc);
  }
  return acc;
}

// Whole WG: 8 waves split [kstart, kstart+klen) for one 16-row strip,
// LDS-reduce, write 16 contiguous results to dst (row-guarded).
__device__ void mv_wg(const float* __restrict__ W, const float* __restrict__ x,
                      int ldw, int row0, int rowmax, int kstart, int klen,
                      float* dst, float (*wred)[16]) {
  const int wave = threadIdx.x >> 5;
  const int lane = threadIdx.x & 31;
  const int kper = klen / NWAVE;
  v8f acc = mv_wave(W, x, ldw, row0, rowmax, kstart + wave * kper, kper);
  // D layout: lane holds rows {0..7}+8*(lane/16) in its 8 regs; all 16
  // columns identical -> lanes 0 and 16 export the 16 row sums.
  if ((lane & 15) == 0) {
    const int half = lane >> 4;
#pragma unroll
    for (int i = 0; i < 8; ++i) wred[wave][half * 8 + i] = acc[i];
  }
  __syncthreads();
  if (threadIdx.x < 16) {
    float s = 0.f;
#pragma unroll
    for (int w = 0; w < NWAVE; ++w) s += wred[w][threadIdx.x];
    if (row0 + (int)threadIdx.x <= rowmax) dst[threadIdx.x] = s;
  }
  __syncthreads();
}

// ---------------- block reduce + LayerNorm (single WG) ----------------
__device__ __forceinline__ float bsum(float v, float* s) {
  s[threadIdx.x] = v;
  __syncthreads();
  for (int o = NTH >> 1; o > 0; o >>= 1) {
    if ((int)threadIdx.x < o) s[threadIdx.x] += s[threadIdx.x + o];
    __syncthreads();
  }
  float r = s[0];
  __syncthreads();
  return r;
}

__device__ void ln_block(const float* __restrict__ xin, const float* __restrict__ w,
                         const float* __restrict__ b, float* out, float* sred) {
  float lv = 0.f;
  for (int i = threadIdx.x; i < Edim; i += NTH) lv += xin[i];
  const float mean = bsum(lv, sred) * (1.f / Edim);
  float vv = 0.f;
  for (int i = threadIdx.x; i < Edim; i += NTH) {
    float c = xin[i] - mean;
    vv += c * c;
  }
  const float var = bsum(vv, sred) * (1.f / Edim);
  const float inv = rsqrtf(var + 1e-5f);
  for (int i = threadIdx.x; i < Edim; i += NTH)
    out[i] = (xin[i] - mean) * inv * w[i] + b[i];
}

// ---------------- the persistent kernel ----------------
__global__ __launch_bounds__(NTH) void rwkv_rnn_kernel(Params P) {
  __shared__ float sred[NTH];
  __shared__ float wred[NWAVE][16];
  const int g = blockIdx.x;
  int ep = 0;
  int* cnt = P.cnt;
  float* ws = P.ws;

  // workspace layout (floats)
  float* x    = ws + 0;        // [E] hidden state
  float* xn   = ws + 1024;     // [E]
  float* xk   = ws + 2048;     // [E]
  float* xv   = ws + 3072;     // [E]
  float* xr   = ws + 4096;     // [E]
  float* prp  = ws + 5120;     // [2][E] Wr partials
  float* pkp  = ws + 7168;     // [2][E]
  float* pvp  = ws + 9216;     // [2][E]
  float* rab  = ws + 11264;    // [E]
  float* pop  = ws + 12288;    // [8][E] Wo partials
  float* xn2  = ws + 20480;    // [E]
  float* fxk  = ws + 21504;    // [E]
  float* fxr  = ws + 22528;    // [E]
  float* frp  = ws + 23552;    // [2][E] Fr partials
  float* kfp  = ws + 25600;    // [H]
  float* rf   = ws + 29696;    // [E]
  float* kf   = ws + 30720;    // [H]
  float* pv2  = ws + 34816;    // [8][E] Fv partials
  float* xfin = ws + 43008;    // [E]

  float* logits = P.out;
  float* oxx = P.out + Vdim;
  float* oaa = oxx + (size_t)Ldim * Edim;
  float* obb = oaa + (size_t)Ldim * Edim;
  float* opp = obb + (size_t)Ldim * Edim;
  float* oxf = opp + (size_t)Ldim * Edim;
  float* oda = oxf + (size_t)Ldim * Edim;
  float* odb = oda + (size_t)Ldim * Edim;

  // ---- setup: x0 = ln0(emb_w[ctx]) ----
  if (g == 0) {
    const int tok = P.ctx[0];
    ln_block(P.emb_w + (size_t)tok * Edim, P.ln0_w, P.ln0_b, x, sred);
  }
  gbar(cnt, ++ep);

  for (int l = 0; l < Ldim; ++l) {
    const size_t lE = (size_t)l * Edim;

    // A: ln1 + token-mix (WG0)
    if (g == 0) {
      ln_block(x, P.ln1_w + lE, P.ln1_b + lE, xn, sred);
      for (int i = threadIdx.x; i < Edim; i += NTH) {
        float n = xn[i], s = P.xx_att[lE + i];
        oxx[lE + i] = n;
        float mk = P.mix_k[lE + i], mv = P.mix_v[lE + i], mr = P.mix_r[lE + i];
        xk[i] = n * mk + s * (1.f - mk);
        xv[i] = n * mv + s * (1.f - mv);
        xr[i] = n * mr + s * (1.f - mr);
      }
    }
    gbar(cnt, ++ep);

    // B: r_pre=Wr@xr, k=Wk@xk, v=Wv@xv  (3 x 64 strips x KPART=2 = 384 WGs)
    if (g < 384) {
      const int mat = g / 128, rem = g % 128;
      const int strip = rem >> 1, p = rem & 1;
      const float* W = (mat == 0 ? P.Wr : mat == 1 ? P.Wk : P.Wv) + lE * Edim;
      const float* xin = (mat == 0 ? xr : mat == 1 ? xk : xv);
      float* part = (mat == 0 ? prp : mat == 1 ? pkp : pvp);
      mv_wg(W, xin, Edim, strip * 16, Edim - 1, p * 512, 512,
            part + p * Edim + strip * 16, wred);
    }
    gbar(cnt, ++ep);

    // C: WKV state update (WG0)
    if (g == 0) {
      for (int i = threadIdx.x; i < Edim; i += NTH) {
        float rp = prp[i] + prp[Edim + i];
        float kk = pkp[i] + pkp[Edim + i];
        float vv = pvp[i] + pvp[Edim + i];
        float r = 1.f / (1.f + expf(-rp));
        float aa = P.aa_att[lE + i], bb = P.bb_att[lE + i], pp = P.pp_att[lE + i];
        float ww = P.tf[lE + i] + kk;
        float q = fmaxf(pp, ww);
        float e1 = expf(pp - q), e2 = expf(ww - q);
        float a = e1 * aa + e2 * vv;
        float b = e1 * bb + e2;
        rab[i] = r * a / b;
        float ww2 = pp + P.td[lE + i];
        float q2 = fmaxf(ww2, kk);
        float f1 = expf(ww2 - q2), f2 = expf(kk - q2);
        oaa[lE + i] = f1 * aa + f2 * vv;
        obb[lE + i] = f1 * bb + f2;
        opp[lE + i] = q2;
        oda[lE + i] = rp;
        odb[lE + i] = r;
      }
    }
    gbar(cnt, ++ep);

    // D: Wo @ (r*a/b)   (64 strips x KPART=8 = 512 WGs)
    {
      const int strip = g >> 3, p = g & 7;
      mv_wg(P.Wo + lE * Edim, rab, Edim, strip * 16, Edim - 1, p * 128, 128,
            pop + p * Edim + strip * 16, wred);
    }
    gbar(cnt, ++ep);

    // E: residual add + ln2 + FFN mix (WG0)
    if (g == 0) {
      for (int i = threadIdx.x; i < Edim; i += NTH) {
        float s = x[i];
#pragma unroll
        for (int p = 0; p < 8; ++p) s += pop[p * Edim + i];
        x[i] = s;
      }
      __syncthreads();
      ln_block(x, P.ln2_w + lE, P.ln2_b + lE, xn2, sred);
      for (int i = threadIdx.x; i < Edim; i += NTH) {
        float n = xn2[i], s = P.xx_ffn[lE + i];
        oxf[lE + i] = n;
        float mk = P.fmk[lE + i], mr = P.fmr[lE + i];
        fxk[i] = n * mk + s * (1.f - mk);
        fxr[i] = n * mr + s * (1.f - mr);
      }
    }
    gbar(cnt, ++ep);

    // F: Fr@fxr (64 strips x KPART=2 = 128 WGs) + Fk@fxk (256 strips, full K)
    if (g < 128) {
      const int strip = g >> 1, p = g & 1;
      mv_wg(P.Fr + lE * Edim, fxr, Edim, strip * 16, Edim - 1, p * 512, 512,
            frp + p * Edim + strip * 16, wred);
    } else if (g < 384) {
      const int strip = g - 128;
      mv_wg(P.Fk + (size_t)l * Hdim * Edim, fxk, Edim, strip * 16, Hdim - 1,
            0, Edim, kfp + strip * 16, wred);
    }
    gbar(cnt, ++ep);

    // G: kf = relu(kf_pre)^2 (WG0..7), rf = sigmoid(sum Fr parts) (WG8)
    if (g < 8) {
      for (int i = g * 512 + threadIdx.x; i < (g + 1) * 512; i += NTH) {
        float t = fmaxf(kfp[i], 0.f);
        kf[i] = t * t;
      }
    } else if (g == 8) {
      for (int i = threadIdx.x; i < Edim; i += NTH) {
        float rp = frp[i] + frp[Edim + i];
        rf[i] = 1.f / (1.f + expf(-rp));
      }
    }
    gbar(cnt, ++ep);

    // H: Fv @ kf  (rows=E, K=H=4096; 64 strips x KPART=8 = 512 WGs)
    {
      const int strip = g >> 3, p = g & 7;
      mv_wg(P.Fv + (size_t)l * Edim * Hdim, kf, Hdim, strip * 16, Edim - 1,
            p * 512, 512, pv2 + p * Edim + strip * 16, wred);
    }
    gbar(cnt, ++ep);

    // I: x += rf * (Fv@kf)  (WG0)
    if (g == 0) {
      for (int i = threadIdx.x; i < Edim; i += NTH) {
        float s = 0.f;
#pragma unroll
        for (int p = 0; p < 8; ++p) s += pv2[p * Edim + i];
        x[i] += rf[i] * s;
      }
    }
    gbar(cnt, ++ep);
  }

  // J: final layernorm (WG0)
  if (g == 0) ln_block(x, P.lno_w, P.lno_b, xfin, sred);
  gbar(cnt, ++ep);

  // K: logits = head_w @ xfin  (3143 strips across all WGs, direct store)
  {
    const int nstrip = (Vdim + 15) / 16;
    for (int s = g; s < nstrip; s += NWG) {
      mv_wg(P.head_w, xfin, Edim, s * 16, Vdim - 1, 0, Edim,
            logits + s * 16, wred);
    }
  }
}

extern "C" void kernel_launch(void* const* d_in, const int* in_sizes, int n_in,
                              void* d_out, int out_size, void* d_ws, size_t ws_size,
                              hipStream_t stream) {
  (void)in_sizes; (void)n_in; (void)out_size; (void)ws_size;
  Params P;
  P.ctx    = (const int*)d_in[0];
  P.xx_att = (const float*)d_in[1];
  P.aa_att = (const float*)d_in[2];
  P.bb_att = (const float*)d_in[3];
  P.pp_att = (const float*)d_in[4];
  P.xx_ffn = (const float*)d_in[5];
  P.emb_w  = (const float*)d_in[6];
  P.head_w = (const float*)d_in[7];
  P.ln0_w  = (const float*)d_in[8];
  P.ln0_b  = (const float*)d_in[9];
  P.ln1_w  = (const float*)d_in[10];
  P.ln1_b  = (const float*)d_in[11];
  P.ln2_w  = (const float*)d_in[12];
  P.ln2_b  = (const float*)d_in[13];
  P.lno_w  = (const float*)d_in[14];
  P.lno_b  = (const float*)d_in[15];
  P.mix_k  = (const float*)d_in[16];
  P.mix_v  = (const float*)d_in[17];
  P.mix_r  = (const float*)d_in[18];
  P.tf     = (const float*)d_in[19];
  P.td     = (const float*)d_in[20];
  P.Wr     = (const float*)d_in[21];
  P.Wk     = (const float*)d_in[22];
  P.Wv     = (const float*)d_in[23];
  P.Wo     = (const float*)d_in[24];
  P.fmk    = (const float*)d_in[25];
  P.fmr    = (const float*)d_in[26];
  P.Fr     = (const float*)d_in[27];
  P.Fk     = (const float*)d_in[28];
  P.Fv     = (const float*)d_in[29];
  P.out    = (float*)d_out;
  P.cnt    = (int*)d_ws;
  P.ws     = (float*)((char*)d_ws + 256);

  // zero the grid-barrier counter region every call (graph-capturable)
  hipMemsetAsync(d_ws, 0, 256, stream);
  rwkv_rnn_kernel<<<NWG, NTH, 0, stream>>>(P);
}